// MultiRangeAttention_16733192585756
// MI455X (gfx1250) — compile-verified
//
#include <hip/hip_runtime.h>

typedef float v2f __attribute__((ext_vector_type(2)));
typedef float v8f __attribute__((ext_vector_type(8)));

#define DIM      192
#define QKV_DIM  576
#define HH       64
#define WW       64
#define HW       4096        // 64*64
#define BATCH    8
#define NTOK     32768       // BATCH*HW
#define HEAD_DIM 32
#define LDS_STRIDE 196       // 192 padded: 196 % 64 != 0 -> conflict-free LDS rows

// ---------------------------------------------------------------------------
// QKV GEMM: qkv[m][n] = sum_k x[b][k][hw] * qkv_w[n][k] + qkv_b[n]
//   m = token = b*4096 + hw.  M=32768, N=576, K=192.
//   block = 256 threads = 8 waves; wave w -> rows [m0+16w, m0+16w+16), cols [n0, n0+16)
// ---------------------------------------------------------------------------
__global__ __launch_bounds__(256)
void qkv_gemm_kernel(const float* __restrict__ x,
                     const float* __restrict__ qkv_w,
                     const float* __restrict__ qkv_b,
                     float* __restrict__ qkv)
{
    __shared__ float wtile[16 * LDS_STRIDE];
    const int n0    = blockIdx.x * 16;
    const int m0blk = blockIdx.y * 128;

    // stage qkv_w[n0:n0+16][0:192] into LDS (shared by all 8 waves)
    for (int idx = threadIdx.x; idx < 16 * DIM; idx += 256) {
        int n = idx / DIM, k = idx - n * DIM;
        wtile[n * LDS_STRIDE + k] = qkv_w[(size_t)(n0 + n) * DIM + k];
    }
    __syncthreads();

    const int lane = threadIdx.x & 31;
    const int wave = threadIdx.x >> 5;
    const int m0   = m0blk + wave * 16;
    const int row  = lane & 15;          // M (and N) index within tile
    const int kp   = (lane >> 4) << 1;   // K pair: 0 or 2

    const int m  = m0 + row;
    const int b  = m >> 12;
    const int hw = m & 4095;
    const float* xa = x + (size_t)b * DIM * HW + hw;   // x[b][k][hw], stride HW in k
    const float* wl = &wtile[row * LDS_STRIDE];

    v8f acc = {};
    #pragma unroll
    for (int k0 = 0; k0 < DIM; k0 += 4) {
        v2f a, bf;
        a.x  = xa[(size_t)(k0 + kp) * HW];
        a.y  = xa[(size_t)(k0 + kp + 1) * HW];
        bf.x = wl[k0 + kp];
        bf.y = wl[k0 + kp + 1];
        acc = __builtin_amdgcn_wmma_f32_16x16x4_f32(false, a, false, bf,
                                                    (short)0, acc, false, false);
    }

    const int   n    = n0 + row;
    const float bias = qkv_b[n];
    const int   mb   = m0 + ((lane >> 4) << 3);   // lanes 16-31 hold M = r+8
    #pragma unroll
    for (int r = 0; r < 8; ++r)
        qkv[(size_t)(mb + r) * QKV_DIM + n] = acc[r] + bias;
}

// ---------------------------------------------------------------------------
// NATTEN branch: one thread per (b, head, h, w); template K/dilation/branch so
// neighbor-index arrays unroll into registers.  Streaming (online) softmax.
// qkv channel layout inside a branch: sel*64 + head*32 + d (sel: 0=q,1=k,2=v)
// ---------------------------------------------------------------------------
template <int K, int D, int BR>
__global__ __launch_bounds__(256)
void natten_kernel(const float* __restrict__ qkv,
                   const float* __restrict__ rpb,
                   float* __restrict__ ycat)
{
    const int gid  = blockIdx.x * 256 + threadIdx.x;   // 65536 threads
    const int hw   = gid & 4095;
    const int t    = gid >> 12;
    const int head = t & 1;
    const int b    = t >> 1;
    const int h    = hw >> 6;
    const int w    = hw & 63;
    constexpr int RK = 2 * K - 1;

    int nh[K], bh[K], nw[K], bw[K];
    {
        const int g = h % D, ii = h / D;
        const int Lg = (HH - g + D - 1) / D;
        int start = ii - K / 2;
        if (start < 0) start = 0;
        if (start > Lg - K) start = Lg - K;
        #pragma unroll
        for (int ki = 0; ki < K; ++ki) { nh[ki] = g + (start + ki) * D; bh[ki] = start + ki - ii + (K - 1); }
    }
    {
        const int g = w % D, ii = w / D;
        const int Lg = (WW - g + D - 1) / D;
        int start = ii - K / 2;
        if (start < 0) start = 0;
        if (start > Lg - K) start = Lg - K;
        #pragma unroll
        for (int ki = 0; ki < K; ++ki) { nw[ki] = g + (start + ki) * D; bw[ki] = start + ki - ii + (K - 1); }
    }

    const size_t tok  = (size_t)b * HW + hw;
    const float* qp   = qkv + tok * QKV_DIM + BR * DIM + head * HEAD_DIM;
    const float  scale = 0.17677669529663687f;   // 32^-0.5

    float4 q[8];
    #pragma unroll
    for (int i = 0; i < 8; ++i) {
        q[i] = *(const float4*)(qp + 4 * i);
        q[i].x *= scale; q[i].y *= scale; q[i].z *= scale; q[i].w *= scale;
    }

    float  mmax = -1e30f, lsum = 0.f;
    float4 acc[8];
    #pragma unroll
    for (int i = 0; i < 8; ++i) acc[i] = make_float4(0.f, 0.f, 0.f, 0.f);

    #pragma unroll
    for (int kh = 0; kh < K; ++kh) {
        const float* rb = rpb + ((size_t)head * RK + bh[kh]) * RK;
        #pragma unroll
        for (int kw = 0; kw < K; ++kw) {
            const size_t nb = ((size_t)b * HW + nh[kh] * 64 + nw[kw]) * QKV_DIM
                            + BR * DIM + head * HEAD_DIM;
            const float* kpn = qkv + nb + 64;    // k
            const float* vpn = qkv + nb + 128;   // v

            float s = 0.f;
            #pragma unroll
            for (int i = 0; i < 8; ++i) {
                float4 kk = *(const float4*)(kpn + 4 * i);
                s += q[i].x * kk.x + q[i].y * kk.y + q[i].z * kk.z + q[i].w * kk.w;
            }
            s += rb[bw[kw]];

            const float mn   = fmaxf(mmax, s);
            const float corr = __expf(mmax - mn);
            const float p    = __expf(s - mn);
            lsum = lsum * corr + p;
            #pragma unroll
            for (int i = 0; i < 8; ++i) {
                float4 vv = *(const float4*)(vpn + 4 * i);
                acc[i].x = acc[i].x * corr + p * vv.x;
                acc[i].y = acc[i].y * corr + p * vv.y;
                acc[i].z = acc[i].z * corr + p * vv.z;
                acc[i].w = acc[i].w * corr + p * vv.w;
            }
            mmax = mn;
        }
    }

    const float inv = 1.f / lsum;
    float* yp = ycat + tok * DIM + BR * 64 + head * 32;
    #pragma unroll
    for (int i = 0; i < 8; ++i) {
        float4 o = make_float4(acc[i].x * inv, acc[i].y * inv, acc[i].z * inv, acc[i].w * inv);
        *(float4*)(yp + 4 * i) = o;
    }
}

// ---------------------------------------------------------------------------
// Output projection GEMM: out[b][n][h][w] = sum_k ycat[m][k]*proj_w[n][k] + proj_b[n]
// ---------------------------------------------------------------------------
__global__ __launch_bounds__(256)
void proj_gemm_kernel(const float* __restrict__ ycat,
                      const float* __restrict__ proj_w,
                      const float* __restrict__ proj_b,
                      float* __restrict__ out)
{
    __shared__ float wtile[16 * LDS_STRIDE];
    const int n0    = blockIdx.x * 16;
    const int m0blk = blockIdx.y * 128;

    for (int idx = threadIdx.x; idx < 16 * DIM; idx += 256) {
        int n = idx / DIM, k = idx - n * DIM;
        wtile[n * LDS_STRIDE + k] = proj_w[(size_t)(n0 + n) * DIM + k];
    }
    __syncthreads();

    const int lane = threadIdx.x & 31;
    const int wave = threadIdx.x >> 5;
    const int m0   = m0blk + wave * 16;
    const int row  = lane & 15;
    const int kp   = (lane >> 4) << 1;

    const float* aa = ycat + (size_t)(m0 + row) * DIM;
    const float* wl = &wtile[row * LDS_STRIDE];

    v8f acc = {};
    #pragma unroll
    for (int k0 = 0; k0 < DIM; k0 += 4) {
        v2f a  = *(const v2f*)(aa + k0 + kp);
        v2f bf = *(const v2f*)(wl + k0 + kp);
        acc = __builtin_amdgcn_wmma_f32_16x16x4_f32(false, a, false, bf,
                                                    (short)0, acc, false, false);
    }

    const int   n    = n0 + row;
    const float bias = proj_b[n];
    const int   mb   = m0 + ((lane >> 4) << 3);
    #pragma unroll
    for (int r = 0; r < 8; ++r) {
        const int m = mb + r;
        const int b = m >> 12, hw = m & 4095;
        out[((size_t)b * DIM + n) * HW + hw] = acc[r] + bias;
    }
}

// ---------------------------------------------------------------------------
extern "C" void kernel_launch(void* const* d_in, const int* in_sizes, int n_in,
                              void* d_out, int out_size, void* d_ws, size_t ws_size,
                              hipStream_t stream)
{
    (void)in_sizes; (void)n_in; (void)out_size; (void)ws_size;
    const float* x      = (const float*)d_in[0];
    const float* qkv_w  = (const float*)d_in[1];
    const float* qkv_b  = (const float*)d_in[2];
    const float* proj_w = (const float*)d_in[3];
    const float* proj_b = (const float*)d_in[4];
    const float* rpb0   = (const float*)d_in[5];
    const float* rpb1   = (const float*)d_in[6];
    const float* rpb2   = (const float*)d_in[7];
    float* out = (float*)d_out;

    float* qkv  = (float*)d_ws;                       // 32768*576 f32 = 75.5 MB
    float* ycat = qkv + (size_t)NTOK * QKV_DIM;       // 32768*192 f32 = 25.2 MB

    // 1) QKV projection (WMMA f32): M=32768, N=576, K=192
    qkv_gemm_kernel<<<dim3(QKV_DIM / 16, NTOK / 128), 256, 0, stream>>>(x, qkv_w, qkv_b, qkv);

    // 2) three NATTEN branches (65536 threads each)
    const int nat_blocks = (BATCH * 2 * HW) / 256;
    natten_kernel<3, 1, 0><<<nat_blocks, 256, 0, stream>>>(qkv, rpb0, ycat);
    natten_kernel<5, 2, 1><<<nat_blocks, 256, 0, stream>>>(qkv, rpb1, ycat);
    natten_kernel<7, 3, 2><<<nat_blocks, 256, 0, stream>>>(qkv, rpb2, ycat);

    // 3) output projection (WMMA f32): M=32768, N=192, K=192; writes NCHW
    proj_gemm_kernel<<<dim3(DIM / 16, NTOK / 128), 256, 0, stream>>>(ycat, proj_w, proj_b, out);
}